// GATConv_3023656976834
// MI455X (gfx1250) — compile-verified
//
#include <hip/hip_runtime.h>
#include <hip/hip_bf16.h>

typedef __attribute__((ext_vector_type(2))) float v2f;
typedef __attribute__((ext_vector_type(8))) float v8f;

#define D 128          // D_IN == D_OUT == 128
#define NEG_SLOPE 0.2f

// ---------------------------------------------------------------------------
// Zero-init kernel for the output accumulator (harness poisons d_out).
// ---------------------------------------------------------------------------
__global__ __launch_bounds__(256) void gat_zero(float* __restrict__ p, long long n) {
    long long i = (long long)blockIdx.x * blockDim.x + threadIdx.x;
    if (i < n) p[i] = 0.0f;
}

// ---------------------------------------------------------------------------
// Fused projection GEMMs via fp32 WMMA:  O[m] = X @ W[m]  for m in {0,1,2}.
// Block = 256 threads (8 waves). blockIdx.x = 16-row tile of X,
// blockIdx.y = which weight matrix. Wave w computes the 16x16 tile at
// columns [16w, 16w+16) using V_WMMA_F32_16X16X4_F32 over K=128 (32 steps).
//
// Fragment layouts (CDNA5 ISA 7.12.2, 32-bit, wave32):
//   A 16x4 : lanes 0-15 -> M=lane, VGPR0=K0,VGPR1=K1 ; lanes 16-31 -> K2,K3
//   B 4x16 : lanes 0-15 -> N=lane, VGPR0=K0,VGPR1=K1 ; lanes 16-31 -> K2,K3
//   C/D    : VGPR v -> row (v + 8*(lane>>4)), col = lane&15
// ---------------------------------------------------------------------------
__global__ __launch_bounds__(256) void gat_gemm_wmma(
    const float* __restrict__ X,
    const float* __restrict__ W0, const float* __restrict__ W1,
    const float* __restrict__ W2,
    float* __restrict__ O0, float* __restrict__ O1, float* __restrict__ O2,
    int n_nodes)
{
    __shared__ float sX[16 * D];   // 8 KB X tile

    const int row0 = blockIdx.x * 16;
    const int m    = blockIdx.y;
    const float* __restrict__ W = (m == 0) ? W0 : (m == 1) ? W1 : W2;
    float* __restrict__       O = (m == 0) ? O0 : (m == 1) ? O1 : O2;

    const int tid = threadIdx.x;

    // Cooperative tile load: 16*128 floats = 512 float4 by 256 threads (2 each).
    {
        float4* Sv = (float4*)sX;
        // float4 index f covers global row row0 + f/32 (clamped for safety).
        int f0 = tid, f1 = tid + 256;
        int r0 = row0 + (f0 >> 5); r0 = r0 < n_nodes ? r0 : n_nodes - 1;
        int r1 = row0 + (f1 >> 5); r1 = r1 < n_nodes ? r1 : n_nodes - 1;
        const float4* X0 = (const float4*)(X + (long long)r0 * D);
        const float4* X1 = (const float4*)(X + (long long)r1 * D);
        Sv[f0] = X0[f0 & 31];
        Sv[f1] = X1[f1 & 31];
    }
    __syncthreads();

    const int wave = tid >> 5;       // 0..7 -> column tile
    const int lane = tid & 31;
    const int half = lane >> 4;      // 0 or 1
    const int lm   = lane & 15;
    const int col0 = wave * 16;

    v8f acc = {};
    #pragma unroll
    for (int k0 = 0; k0 < D; k0 += 4) {
        const int ka = k0 + half * 2;
        v2f a, b;
        a.x = sX[lm * D + ka + 0];
        a.y = sX[lm * D + ka + 1];
        b.x = W[(ka + 0) * D + col0 + lm];
        b.y = W[(ka + 1) * D + col0 + lm];
        // (neg_a, A, neg_b, B, c_mod, C, reuse_a, reuse_b)
        acc = __builtin_amdgcn_wmma_f32_16x16x4_f32(
                false, a, false, b, (short)0, acc, false, false);
    }

    #pragma unroll
    for (int v = 0; v < 8; ++v) {
        const int row = row0 + v + half * 8;
        if (row < n_nodes)
            O[(long long)row * D + col0 + lm] = acc[v];
    }
}

// ---------------------------------------------------------------------------
// Fused SDDMM + LeakyReLU + SpMM scatter. One wave32 per edge:
//   e  = leaky_relu( dot(h_l[src], h_r[dst]) )
//   agg[src] += e * comb[dst]        (fp32 global atomics)
// ---------------------------------------------------------------------------
__global__ __launch_bounds__(256) void gat_edge_fused(
    const float* __restrict__ h_l, const float* __restrict__ h_r,
    const float* __restrict__ comb,
    const int* __restrict__ src, const int* __restrict__ dst,
    float* __restrict__ agg, int n_edges)
{
    const int wavesPerBlock = blockDim.x >> 5;
    const int edge = blockIdx.x * wavesPerBlock + (threadIdx.x >> 5);
    const int lane = threadIdx.x & 31;
    if (edge >= n_edges) return;            // wave-uniform exit

    const int s = src[edge];
    const int d = dst[edge];

    const float4 hl = ((const float4*)(h_l + (long long)s * D))[lane];
    const float4 hr = ((const float4*)(h_r + (long long)d * D))[lane];

    float p = hl.x * hr.x + hl.y * hr.y + hl.z * hr.z + hl.w * hr.w;
    // wave32 butterfly reduction
    #pragma unroll
    for (int off = 16; off > 0; off >>= 1)
        p += __shfl_xor(p, off, 32);

    const float e = (p > 0.0f) ? p : NEG_SLOPE * p;

    const float4 cb = ((const float4*)(comb + (long long)d * D))[lane];
    float* outrow = agg + (long long)s * D + lane * 4;
    unsafeAtomicAdd(outrow + 0, e * cb.x);
    unsafeAtomicAdd(outrow + 1, e * cb.y);
    unsafeAtomicAdd(outrow + 2, e * cb.z);
    unsafeAtomicAdd(outrow + 3, e * cb.w);
}

// ---------------------------------------------------------------------------
// Launch: inputs in setup order: X, W, W_r, W_l, src, dst.
// d_out = agg [n_nodes * 128] fp32.  d_ws holds comb, h_r, h_l.
// ---------------------------------------------------------------------------
extern "C" void kernel_launch(void* const* d_in, const int* in_sizes, int n_in,
                              void* d_out, int out_size, void* d_ws, size_t ws_size,
                              hipStream_t stream) {
    const float* X   = (const float*)d_in[0];
    const float* W   = (const float*)d_in[1];
    const float* W_r = (const float*)d_in[2];
    const float* W_l = (const float*)d_in[3];
    const int*   src = (const int*)d_in[4];
    const int*   dst = (const int*)d_in[5];

    const int n_nodes = in_sizes[0] / D;
    const int n_edges = in_sizes[4];

    const long long per = (long long)n_nodes * D;   // floats per projection
    float* comb = (float*)d_ws;
    float* h_r  = comb + per;
    float* h_l  = h_r + per;
    float* agg  = (float*)d_out;

    // 1) zero output accumulator
    {
        long long n = per;
        int blocks = (int)((n + 255) / 256);
        gat_zero<<<blocks, 256, 0, stream>>>(agg, n);
    }

    // 2) three projection GEMMs (fp32 WMMA); O0=comb(X@W), O1=h_r(X@W_r), O2=h_l(X@W_l)
    {
        dim3 grid((n_nodes + 15) / 16, 3, 1);
        gat_gemm_wmma<<<grid, 256, 0, stream>>>(X, W, W_r, W_l,
                                                comb, h_r, h_l, n_nodes);
    }

    // 3) fused SDDMM + LeakyReLU + scatter-accumulate
    {
        const int wavesPerBlock = 8;                 // 256 threads
        int blocks = (n_edges + wavesPerBlock - 1) / wavesPerBlock;
        gat_edge_fused<<<blocks, 256, 0, stream>>>(h_l, h_r, comb,
                                                   src, dst, agg, n_edges);
    }
}